// DSOM_60447369724283
// MI455X (gfx1250) — compile-verified
//
#include <hip/hip_runtime.h>
#include <math.h>

// Problem constants (from reference)
#define SOM_D    3072          // features per neuron
#define SOM_D4   768           // D / 4 (float4 units)
#define SOM_N    4096          // neurons (64x64 grid)
#define SOM_B    512           // batch (sequential scan steps)
#define SOM_HALF 1536          // D/2, per-lane accumulation span
#define SOM_IT   384           // HALF/4 float4 iterations

typedef __attribute__((ext_vector_type(2))) float v2f;
typedef __attribute__((ext_vector_type(8))) float v8f;

#if defined(__gfx1250__) && __has_builtin(__builtin_amdgcn_global_load_async_to_lds_b128)
#define SOM_ASYNC_LDS 1
// Builtin signature (from clang diagnostic): param 1 is
//   'int __attribute__((vector_size(16))) __device__ *'  (target AS1),
// so cast to exactly that; LDS side is the AS3 twin.
typedef int som_v4i __attribute__((vector_size(16)));
typedef __attribute__((address_space(1))) som_v4i* som_gp4;
typedef __attribute__((address_space(3))) som_v4i* som_lp4;
#else
#define SOM_ASYNC_LDS 0
#endif

// Stage a 12KB x-vector into LDS, one float4 per thread per trip. On gfx1250
// use the async DMA path (GLOBAL_LOAD_ASYNC_TO_LDS_B128, tracked by ASYNCcnt).
__device__ __forceinline__ void som_stage_x(float* __restrict__ dst,
                                            const float* __restrict__ src) {
    const float4* s4 = (const float4*)src;
    float4* d4 = (float4*)dst;
#if SOM_ASYNC_LDS
    for (int i = threadIdx.x; i < SOM_D4; i += 256) {
        __builtin_amdgcn_global_load_async_to_lds_b128(
            (som_gp4)(s4 + i),
            (som_lp4)(d4 + i),
            /*offset=*/0, /*cpol=*/0);
    }
#else
    for (int i = threadIdx.x; i < SOM_D4; i += 256) d4[i] = s4[i];
#endif
}

__device__ __forceinline__ void som_stage_wait() {
#if SOM_ASYNC_LDS
#if __has_builtin(__builtin_amdgcn_s_wait_asynccnt)
    __builtin_amdgcn_s_wait_asynccnt(0);
#else
    asm volatile("s_wait_asynccnt 0x0" ::: "memory");
#endif
#endif
    __syncthreads();
}

// ---------------------------------------------------------------------------
// Init: one u64 argmin slot per step, set to all-ones (max) so atomicMin works.
// ---------------------------------------------------------------------------
__global__ void som_init_slots(unsigned long long* __restrict__ slots) {
    int i = blockIdx.x * blockDim.x + threadIdx.x;
    if (i < SOM_B) slots[i] = ~0ull;
}

// ---------------------------------------------------------------------------
// Fused step kernel (step t):
//   1) apply step t-1's update to the 16 rows this wave owns (register RMW;
//      rows with alpha == 0.0f exactly are untouched -> bit-identical, no store)
//   2) d2[n] = sum_d (x_t[d]-neu[n,d])^2, reduced per-16-row tile by a single
//      v_wmma_f32_16x16x4_f32 against an all-ones B:
//        A[m][0]=partial(m,half0) (lanes 0-15), A[m][2]=partial(m,half1)
//        (lanes 16-31)  =>  D[m][*] = d2[m]
//   3) packed-u64 atomic argmin ((d2_bits<<32)|idx: monotone for d2>=0,
//      ties -> smallest index, matching jnp.argmin)
// ---------------------------------------------------------------------------
__global__ __launch_bounds__(256) void som_step(
        const float* __restrict__ x_cur,
        const float* __restrict__ x_prev,
        float* __restrict__ neu,
        const float* __restrict__ d2_prev,
        float* __restrict__ d2_cur,
        const unsigned long long* __restrict__ slot_prev,
        unsigned long long* __restrict__ slot_cur,
        const float* __restrict__ lr_p,
        const float* __restrict__ e2_p,
        float* __restrict__ bmu_out_prev,   // d_out + 2*(t-1)
        int do_update)
{
    __shared__ float xs[SOM_D];   // x_t
    __shared__ float xp[SOM_D];   // x_{t-1}

    som_stage_x(xs, x_cur);
    if (do_update) som_stage_x(xp, x_prev);
    som_stage_wait();

    const int wave = threadIdx.x >> 5;
    const int lane = threadIdx.x & 31;
    const int tile = blockIdx.x * 8 + wave;   // 16-row tile id, 256 tiles total
    const int m    = lane & 15;               // row within tile
    const int h    = lane >> 4;               // half of the row
    const int row  = tile * 16 + m;

    // alpha for step t-1's update of this row (lane-uniform across halves)
    float alpha = 0.0f;
    if (do_update) {
        const unsigned long long k = *slot_prev;
        const unsigned bmu = (unsigned)k;
        const float bmu_d2 = __uint_as_float((unsigned)(k >> 32));
        const int bx = (int)(bmu >> 6);
        const int by = (int)(bmu & 63u);
        if (blockIdx.x == 0 && threadIdx.x == 0) {
            bmu_out_prev[0] = (float)bx;      // unravel_index(bmu,(64,64))
            bmu_out_prev[1] = (float)by;
        }
        if (bmu_d2 != 0.0f) {
            const float gd = fabsf((float)bx - (float)(row >> 6)) +
                             fabsf((float)by - (float)(row & 63));
            const float neigh = expf(-(gd * gd) / (e2_p[0] * bmu_d2));
            alpha = lr_p[0] * sqrtf(d2_prev[row]) * neigh;
        }
    }

    float4* nr = (float4*)(neu + (size_t)row * SOM_D + h * SOM_HALF);
    const float4* xr  = (const float4*)(xs + h * SOM_HALF);
    const float4* xpr = (const float4*)(xp + h * SOM_HALF);

    float s = 0.0f;
    if (__any(alpha != 0.0f)) {
        // rare path: some row in this wave actually updates
#pragma unroll 4
        for (int i = 0; i < SOM_IT; ++i) {
            float4 nv = nr[i];
            if (alpha != 0.0f) {
                float4 pv = xpr[i];
                nv.x += alpha * (pv.x - nv.x);
                nv.y += alpha * (pv.y - nv.y);
                nv.z += alpha * (pv.z - nv.z);
                nv.w += alpha * (pv.w - nv.w);
                nr[i] = nv;
            }
            float4 xv = xr[i];
            float e0 = xv.x - nv.x, e1 = xv.y - nv.y;
            float e2 = xv.z - nv.z, e3 = xv.w - nv.w;
            s = fmaf(e0, e0, s); s = fmaf(e1, e1, s);
            s = fmaf(e2, e2, s); s = fmaf(e3, e3, s);
        }
    } else {
        // common path: pure distance streaming, deep unroll for MLP
#pragma unroll 8
        for (int i = 0; i < SOM_IT; ++i) {
            float4 nv = nr[i];
            float4 xv = xr[i];
            float e0 = xv.x - nv.x, e1 = xv.y - nv.y;
            float e2 = xv.z - nv.z, e3 = xv.w - nv.w;
            s = fmaf(e0, e0, s); s = fmaf(e1, e1, s);
            s = fmaf(e2, e2, s); s = fmaf(e3, e3, s);
        }
    }

    // WMMA row-sum reduction (paths reconverged: EXEC all-ones here).
    v2f a; a[0] = s;    a[1] = 0.0f;
    v2f b; b[0] = 1.0f; b[1] = 1.0f;
    v8f c = {};
    v8f d = __builtin_amdgcn_wmma_f32_16x16x4_f32(
        false, a, false, b, (short)0, c, false, false);

    // C/D layout: lane<16 VGPR j -> M=j, lane>=16 VGPR j -> M=8+j.
    const int base = tile * 16 + h * 8;
    float best = d[0];
    int   bidx = base;
#pragma unroll
    for (int j = 1; j < 8; ++j) {
        if (d[j] < best) { best = d[j]; bidx = base + j; }
    }

    if (m == 0) { // lanes 0 and 16 publish their 8 rows + argmin candidate
#pragma unroll
        for (int j = 0; j < 8; ++j) d2_cur[base + j] = d[j];
        unsigned long long key =
            ((unsigned long long)__float_as_uint(best) << 32) | (unsigned)bidx;
        atomicMin(slot_cur, key);
    }
}

// ---------------------------------------------------------------------------
// Standalone update (used once, for the final step before the gather).
// Blocks with alpha == 0.0f exactly exit before touching the row.
// ---------------------------------------------------------------------------
__global__ __launch_bounds__(256) void som_update(
        const float* __restrict__ x,
        float* __restrict__ neu,
        const float* __restrict__ d2v,
        const unsigned long long* __restrict__ slot,
        const float* __restrict__ lr_p,
        const float* __restrict__ e2_p,
        float* __restrict__ bmu_out)
{
    const int row = blockIdx.x / 3;
    const int c4  = (blockIdx.x % 3) * 256 + threadIdx.x;

    const unsigned long long k = *slot;
    const unsigned bmu = (unsigned)k;
    const float bmu_d2 = __uint_as_float((unsigned)(k >> 32));
    const int bx = (int)(bmu >> 6);
    const int by = (int)(bmu & 63u);

    if (blockIdx.x == 0 && threadIdx.x == 0) {
        bmu_out[0] = (float)bx;
        bmu_out[1] = (float)by;
    }

    float alpha = 0.0f;
    if (bmu_d2 != 0.0f) {
        const float gd = fabsf((float)bx - (float)(row >> 6)) +
                         fabsf((float)by - (float)(row & 63));
        alpha = lr_p[0] * sqrtf(d2v[row]) *
                expf(-(gd * gd) / (e2_p[0] * bmu_d2));
    }
    if (alpha == 0.0f) return;

    float4* nr = (float4*)neu;
    const float4* x4 = (const float4*)x;
    const int idx = row * SOM_D4 + c4;
    float4 nv = nr[idx];
    float4 xv = x4[c4];
    nv.x += alpha * (xv.x - nv.x);
    nv.y += alpha * (xv.y - nv.y);
    nv.z += alpha * (xv.z - nv.z);
    nv.w += alpha * (xv.w - nv.w);
    nr[idx] = nv;
}

// ---------------------------------------------------------------------------
// Gather: values[t] = final_neurons[bmu_t]
// ---------------------------------------------------------------------------
__global__ __launch_bounds__(256) void som_gather(
        const float* __restrict__ neu,
        const unsigned long long* __restrict__ slots,
        float* __restrict__ out_vals)
{
    const int t = blockIdx.x;
    const unsigned bmu = (unsigned)slots[t];
    const float4* src = (const float4*)(neu + (size_t)bmu * SOM_D);
    float4* dst = (float4*)out_vals + (size_t)t * SOM_D4;
    for (int i = threadIdx.x; i < SOM_D4; i += 256) dst[i] = src[i];
}

// ---------------------------------------------------------------------------
// Host launch. Inputs: [0] input (512,3,32,32) f32, [1] neurons (4096,3072)
// f32, [2] learning_rate f32[1], [3] elasticity_squared f32[1].
// Output: bmu_locs (512,2) then values (512,3,32,32), flattened.
// Workspace: neurons copy (50.33MB) | 512 u64 slots | 2 x 4096 f32 d2.
// ---------------------------------------------------------------------------
extern "C" void kernel_launch(void* const* d_in, const int* in_sizes, int n_in,
                              void* d_out, int out_size, void* d_ws, size_t ws_size,
                              hipStream_t stream) {
    const float* input   = (const float*)d_in[0];
    const float* neurons = (const float*)d_in[1];
    const float* lr      = (const float*)d_in[2];
    const float* e2      = (const float*)d_in[3];
    float* out = (float*)d_out;

    char* ws = (char*)d_ws;
    const size_t neu_bytes = (size_t)SOM_N * SOM_D * sizeof(float);
    float* neu_ws = (float*)ws;
    unsigned long long* slots = (unsigned long long*)(ws + neu_bytes);
    float* d2v = (float*)(ws + neu_bytes + (size_t)SOM_B * sizeof(unsigned long long));

    (void)hipMemcpyAsync(neu_ws, neurons, neu_bytes, hipMemcpyDeviceToDevice, stream);
    som_init_slots<<<2, 256, 0, stream>>>(slots);

    for (int t = 0; t < SOM_B; ++t) {
        const float* xc = input + (size_t)t * SOM_D;
        const float* xp = (t > 0) ? input + (size_t)(t - 1) * SOM_D : xc;
        // step t writes d2 parity (t&1) and slot[t]; reads prev parity/slot
        som_step<<<32, 256, 0, stream>>>(
            xc, xp, neu_ws,
            d2v + ((t + 1) & 1) * SOM_N,          // d2 of step t-1
            d2v + (t & 1) * SOM_N,                // d2 of step t
            (t > 0) ? slots + (t - 1) : slots,    // unread when do_update==0
            slots + t,
            lr, e2,
            (t > 0) ? out + 2 * (t - 1) : out,    // unwritten when do_update==0
            t > 0 ? 1 : 0);
    }
    // final step's update, then the gather from the final table
    som_update<<<SOM_N * 3, 256, 0, stream>>>(
        input + (size_t)(SOM_B - 1) * SOM_D, neu_ws,
        d2v + ((SOM_B - 1) & 1) * SOM_N, slots + (SOM_B - 1),
        lr, e2, out + 2 * (SOM_B - 1));
    som_gather<<<SOM_B, 256, 0, stream>>>(neu_ws, slots, out + 2 * SOM_B);
}